// GraphRoutingLayer_74749610819803
// MI455X (gfx1250) — compile-verified
//
#include <hip/hip_runtime.h>
#include <math.h>

#define BDIM   2
#define NNODES 50000
#define DDIM   64
#define NEDGES 800000
#define EPSLN  1e-5f

typedef __attribute__((ext_vector_type(2))) float v2f;
typedef __attribute__((ext_vector_type(8))) float v8f;

// D = A(16x4 f32) * B(4x16 f32) + C(16x16 f32)
static __device__ __forceinline__ v8f wmma4(v2f a, v2f b, v8f c) {
    return __builtin_amdgcn_wmma_f32_16x16x4_f32(
        /*neg_a=*/false, a, /*neg_b=*/false, b,
        /*c_mod=*/(short)0, c, /*reuse_a=*/false, /*reuse_b=*/false);
}

static __device__ __forceinline__ float gelu_exact(float v) {
    return 0.5f * v * (1.0f + erff(v * 0.70710678118654752f));
}

// ---------------------------------------------------------------------------
// Kernel 1: scatter rf-scaled source features into pre-agg buffer.
// pre[b, dst[e], :] += rf[e] * x[b, src[e], :]
// 16 threads per edge, float4 per thread, both batches.
// x (25.6MB) and pre (25.6MB) both fit in the 192MB L2 -> atomics stay on-chip.
// ---------------------------------------------------------------------------
__global__ void __launch_bounds__(256)
scatter_kernel(const float* __restrict__ x,
               const long long* __restrict__ eidx,
               const float* __restrict__ rf,
               float* __restrict__ pre) {
    int gid = blockIdx.x * blockDim.x + threadIdx.x;
    if (gid >= NEDGES * 16) return;
    int e = gid >> 4;
    int d = (gid & 15) * 4;
    int s = (int)eidx[e];
    int t = (int)eidx[NEDGES + e];
    float r = rf[e];
#pragma unroll
    for (int b = 0; b < BDIM; ++b) {
        const float4 v = *(const float4*)(x + ((size_t)(b * NNODES + s)) * DDIM + d);
        float* p = pre + ((size_t)(b * NNODES + t)) * DDIM + d;
        atomicAdd(p + 0, r * v.x);
        atomicAdd(p + 1, r * v.y);
        atomicAdd(p + 2, r * v.z);
        atomicAdd(p + 3, r * v.w);
    }
}

// ---------------------------------------------------------------------------
// Kernel 2: agg = pre @ W_msg^T via fp32 WMMA, then out = LN1(x + gelu(agg)).
// One wave per 16-node tile; 8 waves per 256-thread block.
// A: pre tile 16x64 (K).  B[k][n] = W_msg[n][k].  4 j-tiles of 16 out-dims.
// ---------------------------------------------------------------------------
__global__ void __launch_bounds__(256)
msg_ln1_kernel(const float* __restrict__ x,
               const float* __restrict__ pre,
               const float* __restrict__ Wm,
               const float* __restrict__ lnw,
               const float* __restrict__ lnb,
               float* __restrict__ out) {
    __shared__ float smem[8][16][72];   // per-wave 16x64 D-tile (+pad)
    const int lane = threadIdx.x & 31;
    const int wave = threadIdx.x >> 5;
    const int TILES = NNODES / 16;      // 3125
    const int tile = blockIdx.x * 8 + wave;
    if (tile >= BDIM * TILES) return;   // wave-uniform: EXEC stays all-ones
    const int b  = tile / TILES;
    const int n0 = (tile - b * TILES) * 16;
    const int lm = lane & 15;
    const int hf = lane >> 4;

    const float* arow = pre + ((size_t)(b * NNODES + n0 + lm)) * DDIM + 2 * hf;

    v8f acc[4] = {};
#pragma unroll
    for (int k0 = 0; k0 < DDIM; k0 += 4) {
        v2f a; a.x = arow[k0]; a.y = arow[k0 + 1];
#pragma unroll
        for (int j = 0; j < 4; ++j) {
            const float* bp = Wm + (size_t)(16 * j + lm) * DDIM + 2 * hf;
            v2f bb; bb.x = bp[k0]; bb.y = bp[k0 + 1];
            acc[j] = wmma4(a, bb, acc[j]);
        }
    }
    // D layout -> LDS: lanes 0-15 rows 0..7, lanes 16-31 rows 8..15, col = 16j+lm
#pragma unroll
    for (int j = 0; j < 4; ++j)
#pragma unroll
        for (int r = 0; r < 8; ++r)
            smem[wave][8 * hf + r][16 * j + lm] = acc[j][r];

    // LayerNorm: 2 lanes per node, 32 dims each, shfl_xor(1) to combine
    const int node = lane >> 1;
    const int d0 = (lane & 1) * 32;
    const float* xrow = x + ((size_t)(b * NNODES + n0 + node)) * DDIM;
    float hbuf[32];
    float sum = 0.f;
#pragma unroll
    for (int i = 0; i < 32; ++i) {
        float h = xrow[d0 + i] + gelu_exact(smem[wave][node][d0 + i]);
        hbuf[i] = h;
        sum += h;
    }
    sum += __shfl_xor(sum, 1, 32);
    float mean = sum * (1.0f / 64.0f);
    float vs = 0.f;
#pragma unroll
    for (int i = 0; i < 32; ++i) { float dl = hbuf[i] - mean; vs += dl * dl; }
    vs += __shfl_xor(vs, 1, 32);
    float rstd = rsqrtf(vs * (1.0f / 64.0f) + EPSLN);
    float* orow = out + ((size_t)(b * NNODES + n0 + node)) * DDIM;
#pragma unroll
    for (int i = 0; i < 32; ++i) {
        int d = d0 + i;
        orow[d] = (hbuf[i] - mean) * rstd * lnw[d] + lnb[d];
    }
}

// ---------------------------------------------------------------------------
// Kernel 3: h = gelu(out @ W1^T + b1) [16x128 in LDS], mlp = h @ W2^T + b2,
// y = LN2(out + mlp).  One wave per 16-node tile; 4 waves per 128-thr block.
// ---------------------------------------------------------------------------
__global__ void __launch_bounds__(128)
mlp_ln2_kernel(const float* __restrict__ outb,
               const float* __restrict__ W1,
               const float* __restrict__ b1,
               const float* __restrict__ W2,
               const float* __restrict__ b2,
               const float* __restrict__ lnw,
               const float* __restrict__ lnb,
               float* __restrict__ y) {
    __shared__ float hs[4][16][132];    // per-wave 16x128 h-tile (+pad)
    const int lane = threadIdx.x & 31;
    const int wave = threadIdx.x >> 5;
    const int TILES = NNODES / 16;
    const int tile = blockIdx.x * 4 + wave;
    if (tile >= BDIM * TILES) return;   // wave-uniform
    const int b  = tile / TILES;
    const int n0 = (tile - b * TILES) * 16;
    const int lm = lane & 15;
    const int hf = lane >> 4;

    const float* arow = outb + ((size_t)(b * NNODES + n0 + lm)) * DDIM + 2 * hf;

    // GEMM1: [16x64] x [64x128] -> 8 j-tiles
    v8f acc[8] = {};
#pragma unroll
    for (int k0 = 0; k0 < DDIM; k0 += 4) {
        v2f a; a.x = arow[k0]; a.y = arow[k0 + 1];
#pragma unroll
        for (int j = 0; j < 8; ++j) {
            const float* bp = W1 + (size_t)(16 * j + lm) * DDIM + 2 * hf;
            v2f bb; bb.x = bp[k0]; bb.y = bp[k0 + 1];
            acc[j] = wmma4(a, bb, acc[j]);
        }
    }
    // epilogue: h = gelu(acc + b1[col]) -> LDS
#pragma unroll
    for (int j = 0; j < 8; ++j) {
        float bias = b1[16 * j + lm];
#pragma unroll
        for (int r = 0; r < 8; ++r)
            hs[wave][8 * hf + r][16 * j + lm] = gelu_exact(acc[j][r] + bias);
    }

    // GEMM2: [16x128] x [128x64] -> 4 j-tiles, A from LDS
    v8f acc2[4] = {};
#pragma unroll
    for (int k0 = 0; k0 < 2 * DDIM; k0 += 4) {
        v2f a; a.x = hs[wave][lm][k0 + 2 * hf];
        a.y = hs[wave][lm][k0 + 2 * hf + 1];
#pragma unroll
        for (int j = 0; j < 4; ++j) {
            const float* bp = W2 + (size_t)(16 * j + lm) * (2 * DDIM) + 2 * hf;
            v2f bb; bb.x = bp[k0]; bb.y = bp[k0 + 1];
            acc2[j] = wmma4(a, bb, acc2[j]);
        }
    }
    // mlp tile -> LDS (reuse cols 0..63 of h-tile; all h reads precede in
    // program order and same-wave DS ops are in-order)
#pragma unroll
    for (int j = 0; j < 4; ++j)
#pragma unroll
        for (int r = 0; r < 8; ++r)
            hs[wave][8 * hf + r][16 * j + lm] = acc2[j][r];

    // LN2 over (out + mlp + b2)
    const int node = lane >> 1;
    const int d0 = (lane & 1) * 32;
    const float* orow = outb + ((size_t)(b * NNODES + n0 + node)) * DDIM;
    float hbuf[32];
    float sum = 0.f;
#pragma unroll
    for (int i = 0; i < 32; ++i) {
        int d = d0 + i;
        float t = orow[d] + hs[wave][node][d] + b2[d];
        hbuf[i] = t;
        sum += t;
    }
    sum += __shfl_xor(sum, 1, 32);
    float mean = sum * (1.0f / 64.0f);
    float vs = 0.f;
#pragma unroll
    for (int i = 0; i < 32; ++i) { float dl = hbuf[i] - mean; vs += dl * dl; }
    vs += __shfl_xor(vs, 1, 32);
    float rstd = rsqrtf(vs * (1.0f / 64.0f) + EPSLN);
    float* yrow = y + ((size_t)(b * NNODES + n0 + node)) * DDIM;
#pragma unroll
    for (int i = 0; i < 32; ++i) {
        int d = d0 + i;
        yrow[d] = (hbuf[i] - mean) * rstd * lnw[d] + lnb[d];
    }
}

// ---------------------------------------------------------------------------
extern "C" void kernel_launch(void* const* d_in, const int* in_sizes, int n_in,
                              void* d_out, int out_size, void* d_ws, size_t ws_size,
                              hipStream_t stream) {
    const float*     x    = (const float*)d_in[0];
    const long long* eidx = (const long long*)d_in[1];   // int64 [2, E]
    const float*     rf   = (const float*)d_in[2];
    const float*     Wm   = (const float*)d_in[3];
    const float*     ln1w = (const float*)d_in[4];
    const float*     ln1b = (const float*)d_in[5];
    const float*     ln2w = (const float*)d_in[6];
    const float*     ln2b = (const float*)d_in[7];
    const float*     W1   = (const float*)d_in[8];
    const float*     b1   = (const float*)d_in[9];
    const float*     W2   = (const float*)d_in[10];
    const float*     b2   = (const float*)d_in[11];
    float* yout = (float*)d_out;

    float* pre  = (float*)d_ws;                          // [B,N,D] pre-agg
    float* outb = pre + (size_t)BDIM * NNODES * DDIM;    // [B,N,D] post-LN1

    hipMemsetAsync(pre, 0, sizeof(float) * (size_t)BDIM * NNODES * DDIM, stream);

    {
        int total = NEDGES * 16;
        scatter_kernel<<<(total + 255) / 256, 256, 0, stream>>>(x, eidx, rf, pre);
    }
    {
        int tiles = BDIM * (NNODES / 16);                // 6250
        msg_ln1_kernel<<<(tiles + 7) / 8, 256, 0, stream>>>(x, pre, Wm, ln1w, ln1b, outb);
        mlp_ln2_kernel<<<(tiles + 3) / 4, 128, 0, stream>>>(outb, W1, b1, W2, b2, ln2w, ln2b, yout);
    }
}